// HGNN_86045374808535
// MI455X (gfx1250) — compile-verified
//
#include <hip/hip_runtime.h>

// HGNN forward for MI455X (gfx1250, wave32).
// - fp32 WMMA (V_WMMA_F32_16X16X4_F32) for dense GEMMs, 16x64 output per wave.
// - CSR-gather segment sums (no feature-space atomics; atomic-throughput-bound
//   scatter replaced by bandwidth-bound gather; working set fits 192MB L2).

constexpr int   N_NODES   = 50000;
constexpr int   N_EDGES   = 10000;
constexpr int   NNZ       = 400000;
constexpr int   D         = 256;
constexpr float NEG_SLOPE = 0.1f;

typedef float v2f __attribute__((ext_vector_type(2)));
typedef float v8f __attribute__((ext_vector_type(8)));

// ---------------------------------------------------------------- utilities
__global__ void zero_i32(int* __restrict__ p, int n) {
  int i = blockIdx.x * blockDim.x + threadIdx.x;
  if (i < n) p[i] = 0;
}

// ------------------------------------------------------------- degrees/coeff
__global__ void degree_kernel(const int* __restrict__ ni, const int* __restrict__ ei,
                              int* __restrict__ dV, int* __restrict__ dE) {
  int i = blockIdx.x * blockDim.x + threadIdx.x;
  if (i < NNZ) {
    atomicAdd(dV + ni[i], 1);
    atomicAdd(dE + ei[i], 1);
  }
}

__global__ void coeff_kernel(const int* __restrict__ ni, const int* __restrict__ ei,
                             const int* __restrict__ dV, const int* __restrict__ dE,
                             float* __restrict__ coeff) {
  int i = blockIdx.x * blockDim.x + threadIdx.x;
  if (i < NNZ) {
    coeff[i] = rsqrtf((float)dV[ni[i]]) / (float)dE[ei[i]];
  }
}

// ------------------------------------------------------ exclusive scan (int)
// Scans m = n_in+1 virtual elements: val(g) = g < n_in ? in[g] : 0, so
// out[n_in] ends up holding the total (== NNZ).
__global__ void scan_block_excl(const int* __restrict__ in, int n_in,
                                int* __restrict__ out, int* __restrict__ bsums) {
  __shared__ int sh[256];
  const int m   = n_in + 1;
  const int tid = threadIdx.x;
  const int gid = blockIdx.x * 256 + tid;
  int v = (gid < n_in) ? in[gid] : 0;
  sh[tid] = v;
  __syncthreads();
  for (int off = 1; off < 256; off <<= 1) {
    int t = (tid >= off) ? sh[tid - off] : 0;
    __syncthreads();
    sh[tid] += t;
    __syncthreads();
  }
  if (gid < m) out[gid] = sh[tid] - v;      // exclusive
  if (tid == 255) bsums[blockIdx.x] = sh[255];
}

__global__ void scan_single_excl(int* __restrict__ data, int n) {  // n <= 256
  __shared__ int sh[256];
  const int tid = threadIdx.x;
  int v = (tid < n) ? data[tid] : 0;
  sh[tid] = v;
  __syncthreads();
  for (int off = 1; off < 256; off <<= 1) {
    int t = (tid >= off) ? sh[tid - off] : 0;
    __syncthreads();
    sh[tid] += t;
    __syncthreads();
  }
  if (tid < n) data[tid] = sh[tid] - v;
}

__global__ void scan_add(int* __restrict__ out, const int* __restrict__ bsums, int m) {
  int gid = blockIdx.x * 256 + threadIdx.x;
  if (gid < m) out[gid] += bsums[blockIdx.x];
}

// --------------------------------------------------------------- CSR build
__global__ void csr_fill(const int* __restrict__ ni, const int* __restrict__ ei,
                         const float* __restrict__ coeff,
                         const int* __restrict__ e_offs, int* __restrict__ e_cnt,
                         int* __restrict__ e_adj, float* __restrict__ e_ac,
                         const int* __restrict__ n_offs, int* __restrict__ n_cnt,
                         int* __restrict__ n_adj, float* __restrict__ n_ac) {
  int i = blockIdx.x * blockDim.x + threadIdx.x;
  if (i < NNZ) {
    const int   n = ni[i], e = ei[i];
    const float c = coeff[i];
    int pe = e_offs[e] + atomicAdd(e_cnt + e, 1);
    e_adj[pe] = n;  e_ac[pe] = c;
    int pn = n_offs[n] + atomicAdd(n_cnt + n, 1);
    n_adj[pn] = e;  n_ac[pn] = c;
  }
}

// -------------------------------------------------------------- WMMA GEMM
// Y[N,256] = X[N,256] @ W[256,256]^T + bias. One wave computes a 16x64 output
// strip (4 accumulators, A fragment reused 4x). K-loop: 64 x WMMA f32 16x16x4.
__global__ void __launch_bounds__(128)
gemm_bias_wmma(const float* __restrict__ X, const float* __restrict__ W,
               const float* __restrict__ bias, float* __restrict__ Y) {
  const int wave = blockIdx.x * 4 + (threadIdx.x >> 5);   // 0..12499
  const int lane = threadIdx.x & 31;
  const int rt   = wave >> 2;        // row tile     (0..3124)
  const int cg   = wave & 3;         // column group (cols cg*64 .. cg*64+63)
  const int half = lane >> 4;
  const int r    = lane & 15;

  const v2f* __restrict__ ap = (const v2f*)(X + (size_t)(rt * 16 + r) * D);
  const float* wbase = W + (size_t)(cg * 64 + r) * D;
  const v2f* __restrict__ bp0 = (const v2f*)(wbase);
  const v2f* __restrict__ bp1 = (const v2f*)(wbase + 16 * D);
  const v2f* __restrict__ bp2 = (const v2f*)(wbase + 32 * D);
  const v2f* __restrict__ bp3 = (const v2f*)(wbase + 48 * D);

  v8f c0 = {}, c1 = {}, c2 = {}, c3 = {};
#pragma unroll 4
  for (int k = 0; k < D; k += 4) {
    const int idx = (k >> 1) + half;
    v2f a  = ap[idx];
    v2f b0 = bp0[idx], b1 = bp1[idx], b2 = bp2[idx], b3 = bp3[idx];
    c0 = __builtin_amdgcn_wmma_f32_16x16x4_f32(false, a, false, b0, (short)0, c0, false, false);
    c1 = __builtin_amdgcn_wmma_f32_16x16x4_f32(false, a, false, b1, (short)0, c1, false, false);
    c2 = __builtin_amdgcn_wmma_f32_16x16x4_f32(false, a, false, b2, (short)0, c2, false, false);
    c3 = __builtin_amdgcn_wmma_f32_16x16x4_f32(false, a, false, b3, (short)0, c3, false, false);
  }

  // C/D layout: VGPR j, lanes 0-15 -> M=j, lanes 16-31 -> M=j+8; N = lane%16.
  const int col0 = cg * 64 + r;
  float* __restrict__ obase = Y + (size_t)(rt * 16 + half * 8) * D;
  const float bv0 = bias[col0], bv1 = bias[col0 + 16],
              bv2 = bias[col0 + 32], bv3 = bias[col0 + 48];
#pragma unroll
  for (int j = 0; j < 8; ++j) {
    float* row = obase + (size_t)j * D + col0;
    row[0]  = c0[j] + bv0;
    row[16] = c1[j] + bv1;
    row[32] = c2[j] + bv2;
    row[48] = c3[j] + bv3;
  }
}

// ----------------------------------------------- fused CSR segment gather
// dst[seg, t] = act( (skip ? skip[seg,t] : 0) + sum_j ac[j]*src[adj[j], t] )
// One 256-thread block per segment; adjacency staged through LDS in chunks.
__global__ void __launch_bounds__(256)
gather_segment(const float* __restrict__ src, const float* __restrict__ skip,
               float* __restrict__ dst, const int* __restrict__ offs,
               const int* __restrict__ adj, const float* __restrict__ ac,
               int do_leaky) {
  __shared__ int   s_idx[128];
  __shared__ float s_c[128];
  const int seg = blockIdx.x;
  const int t   = threadIdx.x;                 // feature index 0..255
  const int s0  = offs[seg], s1 = offs[seg + 1];

  float acc = skip ? skip[(size_t)seg * D + t] : 0.0f;
  for (int base = s0; base < s1; base += 128) {
    const int cnt = min(128, s1 - base);
    if (t < cnt) { s_idx[t] = adj[base + t]; s_c[t] = ac[base + t]; }
    __syncthreads();
    for (int j = 0; j < cnt; ++j)
      acc += s_c[j] * src[(size_t)s_idx[j] * D + t];
    __syncthreads();
  }
  if (do_leaky) acc = acc >= 0.0f ? acc : NEG_SLOPE * acc;
  dst[(size_t)seg * D + t] = acc;
}

// ------------------------------------------------------------------- launch
extern "C" void kernel_launch(void* const* d_in, const int* in_sizes, int n_in,
                              void* d_out, int out_size, void* d_ws, size_t ws_size,
                              hipStream_t stream) {
  (void)in_sizes; (void)n_in; (void)out_size; (void)ws_size;

  const float* X_in     = (const float*)d_in[0];
  const int*   node_ids = (const int*)d_in[1];
  const int*   edge_ids = (const int*)d_in[2];
  // d_in[3] = n_hyperedges (compile-time constant here)
  const float* fc1_w   = (const float*)d_in[4];
  const float* fc1_b   = (const float*)d_in[5];
  const float* proj1_w = (const float*)d_in[6];
  const float* proj1_b = (const float*)d_in[7];
  const float* fc2_w   = (const float*)d_in[8];
  const float* fc2_b   = (const float*)d_in[9];
  const float* proj2_w = (const float*)d_in[10];
  const float* proj2_b = (const float*)d_in[11];

  // d_out = [E_final (10000*256) | X_final (50000*256)]
  float* Eout = (float*)d_out;
  float* Xcur = (float*)d_out + (size_t)N_EDGES * D;

  // Workspace carve-up (all 4-byte elements)
  char* wsb = (char*)d_ws;
  auto take = [&](size_t elems) { void* p = wsb; wsb += elems * 4; return p; };
  int*   dVi    = (int*)  take(N_NODES);
  int*   dEi    = (int*)  take(N_EDGES);
  float* coeff  = (float*)take(NNZ);
  int*   e_offs = (int*)  take(N_EDGES + 1);
  int*   n_offs = (int*)  take(N_NODES + 1);
  int*   e_cnt  = (int*)  take(N_EDGES);
  int*   n_cnt  = (int*)  take(N_NODES);
  int*   e_adj  = (int*)  take(NNZ);
  float* e_ac   = (float*)take(NNZ);
  int*   n_adj  = (int*)  take(NNZ);
  float* n_ac   = (float*)take(NNZ);
  int*   bsums  = (int*)  take(256);
  float* S      = (float*)take((size_t)N_NODES * D);   // skip buffer
  float* T      = (float*)take((size_t)N_NODES * D);   // fc output
  float* Eb     = (float*)take((size_t)N_EDGES * D);   // edge features

  const int TPB = 256;
  const int ND  = N_NODES * D;
  const int gemm_blocks = (N_NODES / 16) * 4 / 4;      // 3125 (wave per 16x64 strip)
  const int nnzg = (NNZ + TPB - 1) / TPB;

  // ---- degrees + normalization coefficients
  zero_i32<<<(N_NODES + TPB - 1) / TPB, TPB, 0, stream>>>(dVi, N_NODES);
  zero_i32<<<(N_EDGES + TPB - 1) / TPB, TPB, 0, stream>>>(dEi, N_EDGES);
  degree_kernel<<<nnzg, TPB, 0, stream>>>(node_ids, edge_ids, dVi, dEi);
  coeff_kernel<<<nnzg, TPB, 0, stream>>>(node_ids, edge_ids, dVi, dEi, coeff);

  // ---- exclusive scans -> CSR row offsets (length n+1, last = NNZ)
  {
    const int mE = N_EDGES + 1, nbE = (mE + 255) / 256;
    scan_block_excl<<<nbE, 256, 0, stream>>>(dEi, N_EDGES, e_offs, bsums);
    scan_single_excl<<<1, 256, 0, stream>>>(bsums, nbE);
    scan_add<<<nbE, 256, 0, stream>>>(e_offs, bsums, mE);
    const int mN = N_NODES + 1, nbN = (mN + 255) / 256;
    scan_block_excl<<<nbN, 256, 0, stream>>>(dVi, N_NODES, n_offs, bsums);
    scan_single_excl<<<1, 256, 0, stream>>>(bsums, nbN);
    scan_add<<<nbN, 256, 0, stream>>>(n_offs, bsums, mN);
  }

  // ---- CSR adjacency fill
  zero_i32<<<(N_EDGES + TPB - 1) / TPB, TPB, 0, stream>>>(e_cnt, N_EDGES);
  zero_i32<<<(N_NODES + TPB - 1) / TPB, TPB, 0, stream>>>(n_cnt, N_NODES);
  csr_fill<<<nnzg, TPB, 0, stream>>>(node_ids, edge_ids, coeff,
                                     e_offs, e_cnt, e_adj, e_ac,
                                     n_offs, n_cnt, n_adj, n_ac);

  // ---- X working copy lives in d_out's X region
  hipMemcpyAsync(Xcur, X_in, (size_t)ND * sizeof(float), hipMemcpyDeviceToDevice, stream);

  auto half_layer = [&](const float* pW, const float* pB, const float* fW, const float* fB) {
    gemm_bias_wmma<<<gemm_blocks, 128, 0, stream>>>(Xcur, pW, pB, S);   // skip path
    gemm_bias_wmma<<<gemm_blocks, 128, 0, stream>>>(Xcur, fW, fB, T);   // fc path
    // E = leaky(node2edge(T))
    gather_segment<<<N_EDGES, TPB, 0, stream>>>(T, nullptr, Eb, e_offs, e_adj, e_ac, 1);
    // X = leaky(edge2node(E) + skip)
    gather_segment<<<N_NODES, TPB, 0, stream>>>(Eb, S, Xcur, n_offs, n_adj, n_ac, 1);
  };

  for (int i = 0; i < 2; ++i) {
    const size_t wo = (size_t)i * D * D, bo = (size_t)i * D;
    half_layer(proj1_w + wo, proj1_b + bo, fc1_w + wo, fc1_b + bo);
    half_layer(proj2_w + wo, proj2_b + bo, fc2_w + wo, fc2_b + bo);
  }

  // ---- E_final = node2edge(X)  (no activation)
  gather_segment<<<N_EDGES, TPB, 0, stream>>>(Xcur, nullptr, Eout, e_offs, e_adj, e_ac, 0);
}